// CNN_LSTM_46179488366928
// MI455X (gfx1250) — compile-verified
//
#include <hip/hip_runtime.h>
#include <hip/hip_bf16.h>

typedef __attribute__((ext_vector_type(2))) float v2f;
typedef __attribute__((ext_vector_type(8))) float v8f;

#define LSTM_H 5
#define GATES  20      // 4*H
#define DIN    225
#define KPAD   229     // padded LDS row stride (odd -> conflict free)

__device__ __forceinline__ float sigmoid_f(float x) {
  return 1.0f / (1.0f + __expf(-x));     // v_exp_f32 + rcp
}

__device__ __forceinline__ float tanh_f(float x) {
#if __has_builtin(__builtin_amdgcn_tanhf)
  return __builtin_amdgcn_tanhf(x);      // gfx1250 v_tanh_f32 (TRANS op)
#else
  float e = __expf(2.0f * x);
  return 1.0f - 2.0f / (e + 1.0f);
#endif
}

// ---------------------------------------------------------------------------
// Kernel A: xg0[m, 0..19] = x[m, :] @ Wih0^T + (bih0 + bhh0),  m = b*T + t
// One wave per 16-row M-tile; two WMMA N-tiles cover the 20 gate columns.
// K = 225 = 56*4 (WMMA f32 16x16x4) + 1 scalar remainder column.
// A loads use default temporal policy: each 128B line of x feeds 8 K-steps.
// ---------------------------------------------------------------------------
__global__ void __launch_bounds__(256)
xg_gemm_kernel(const float* __restrict__ x, const float* __restrict__ Wih,
               const float* __restrict__ bih, const float* __restrict__ bhh,
               float* __restrict__ xg) {
  __shared__ float sW[32 * KPAD];   // rows n=0..31 (n>=20 zero), k=0..224
  __shared__ float sBias[32];

  for (int idx = threadIdx.x; idx < 32 * DIN; idx += 256) {
    int n = idx / DIN, k = idx % DIN;
    sW[n * KPAD + k] = (n < GATES) ? Wih[n * DIN + k] : 0.0f;
  }
  if (threadIdx.x < 32)
    sBias[threadIdx.x] =
        (threadIdx.x < GATES) ? (bih[threadIdx.x] + bhh[threadIdx.x]) : 0.0f;
  __syncthreads();

  const int wave = threadIdx.x >> 5;
  const int lane = threadIdx.x & 31;
  const int lm   = lane & 15;       // M (for A) / N (for B,C) within tile
  const int hi   = lane >> 4;       // lane half
  const int lk   = hi * 2;          // K sub-offset: lanes 0-15 -> K {0,1}, 16-31 -> K {2,3}
  const long row0 = ((long)blockIdx.x * 8 + wave) * 16;

  const float* __restrict__ arow = x + (row0 + lm) * DIN;
  const float* __restrict__ w0 = sW + lm * KPAD;          // N-tile 0: cols 0..15
  const float* __restrict__ w1 = sW + (16 + lm) * KPAD;   // N-tile 1: cols 16..31

  v8f c0 = {0.f, 0.f, 0.f, 0.f, 0.f, 0.f, 0.f, 0.f};
  v8f c1 = {0.f, 0.f, 0.f, 0.f, 0.f, 0.f, 0.f, 0.f};

  for (int k0 = 0; k0 < DIN - 1; k0 += 4) {   // K = 0..223
    v2f a, b0, b1;
    a.x  = arow[k0 + lk];
    a.y  = arow[k0 + lk + 1];
    b0.x = w0[k0 + lk];  b0.y = w0[k0 + lk + 1];
    b1.x = w1[k0 + lk];  b1.y = w1[k0 + lk + 1];
    c0 = __builtin_amdgcn_wmma_f32_16x16x4_f32(false, a, false, b0, (short)0, c0, false, false);
    c1 = __builtin_amdgcn_wmma_f32_16x16x4_f32(false, a, false, b1, (short)0, c1, false, false);
  }

  // Remainder column K=224, bias, and stores.
  // C/D layout: vgpr r, lanes 0-15 -> (M=r, N=lm); lanes 16-31 -> (M=r+8, N=lm).
  const float wn0 = w0[DIN - 1];
  const float wn1 = w1[DIN - 1];
  const float bb0 = sBias[lm];
  const float bb1 = sBias[16 + lm];
#pragma unroll
  for (int r = 0; r < 8; ++r) {
    const long m  = row0 + r + hi * 8;
    const float xv = x[m * DIN + (DIN - 1)];
    xg[m * GATES + lm] = c0[r] + xv * wn0 + bb0;
    if (lm < GATES - 16)
      xg[m * GATES + 16 + lm] = c1[r] + xv * wn1 + bb1;
  }
}

// ---------------------------------------------------------------------------
// Kernel B: fused 4-layer LSTM recurrence + fc1(ReLU) + fc2(sigmoid).
// One thread per batch element (512 independent chains); all state in VGPRs.
// Weights are wave-uniform with constant indices -> compiler lowers them to
// s_load through the scalar/constant cache (co-issues with VALU) and keeps
// the hot ones in SGPRs as v_fmac scalar operands. Torch gate order: i,f,g,o.
// ---------------------------------------------------------------------------
__device__ __forceinline__ void lstm_cell(const float* __restrict__ Wih,
                                          const float* __restrict__ Whh,
                                          const float* __restrict__ bih,
                                          const float* __restrict__ bhh,
                                          const float* in5, float* h, float* c) {
  float g[GATES];
#pragma unroll
  for (int j = 0; j < GATES; ++j) {
    float a = bih[j] + bhh[j];
#pragma unroll
    for (int k = 0; k < LSTM_H; ++k)
      a += Wih[j * LSTM_H + k] * in5[k] + Whh[j * LSTM_H + k] * h[k];
    g[j] = a;
  }
#pragma unroll
  for (int k = 0; k < LSTM_H; ++k) {
    float i  = sigmoid_f(g[k]);
    float f  = sigmoid_f(g[LSTM_H + k]);
    float gg = tanh_f(g[2 * LSTM_H + k]);
    float o  = sigmoid_f(g[3 * LSTM_H + k]);
    float cn = f * c[k] + i * gg;
    c[k] = cn;
    h[k] = o * tanh_f(cn);
  }
}

__global__ void __launch_bounds__(256)
lstm_chain_kernel(const float* __restrict__ xg0,
                  const float* __restrict__ Whh0,
                  const float* __restrict__ Wih1, const float* __restrict__ Whh1,
                  const float* __restrict__ bih1, const float* __restrict__ bhh1,
                  const float* __restrict__ fc1W, const float* __restrict__ fc1b,
                  const float* __restrict__ Wih2, const float* __restrict__ Whh2,
                  const float* __restrict__ bih2, const float* __restrict__ bhh2,
                  const float* __restrict__ Wih3, const float* __restrict__ Whh3,
                  const float* __restrict__ bih3, const float* __restrict__ bhh3,
                  const float* __restrict__ fc2W, const float* __restrict__ fc2b,
                  float* __restrict__ out, int Bsz, int T) {
  const int b = blockIdx.x * blockDim.x + threadIdx.x;
  if (b >= Bsz) return;

  float h0[LSTM_H] = {}, c0[LSTM_H] = {};
  float h1[LSTM_H] = {}, c1[LSTM_H] = {};
  float h2[LSTM_H] = {}, c2[LSTM_H] = {};
  float h3[LSTM_H] = {}, c3[LSTM_H] = {};

  // 20-float rows, 80B stride -> every row is 16B aligned: 5 x b128 loads.
  const float4* __restrict__ xq =
      (const float4*)(xg0 + (size_t)b * T * GATES);
  float* orow = out + (size_t)b * T * LSTM_H;

  for (int t = 0; t < T; ++t) {
    float xrow[GATES];
    {
      float4 q0 = xq[0], q1 = xq[1], q2 = xq[2], q3 = xq[3], q4 = xq[4];
      xrow[0]  = q0.x; xrow[1]  = q0.y; xrow[2]  = q0.z; xrow[3]  = q0.w;
      xrow[4]  = q1.x; xrow[5]  = q1.y; xrow[6]  = q1.z; xrow[7]  = q1.w;
      xrow[8]  = q2.x; xrow[9]  = q2.y; xrow[10] = q2.z; xrow[11] = q2.w;
      xrow[12] = q3.x; xrow[13] = q3.y; xrow[14] = q3.z; xrow[15] = q3.w;
      xrow[16] = q4.x; xrow[17] = q4.y; xrow[18] = q4.z; xrow[19] = q4.w;
    }
    // --- layer 0: input projection precomputed in xg0 ---
    {
      float g[GATES];
#pragma unroll
      for (int j = 0; j < GATES; ++j) {
        float a = xrow[j];
#pragma unroll
        for (int k = 0; k < LSTM_H; ++k) a += Whh0[j * LSTM_H + k] * h0[k];
        g[j] = a;
      }
#pragma unroll
      for (int k = 0; k < LSTM_H; ++k) {
        float i  = sigmoid_f(g[k]);
        float f  = sigmoid_f(g[LSTM_H + k]);
        float gg = tanh_f(g[2 * LSTM_H + k]);
        float o  = sigmoid_f(g[3 * LSTM_H + k]);
        float cn = f * c0[k] + i * gg;
        c0[k] = cn;
        h0[k] = o * tanh_f(cn);
      }
    }
    // --- layer 1 ---
    lstm_cell(Wih1, Whh1, bih1, bhh1, h0, h1, c1);
    // --- fc1 + ReLU ---
    float r1[LSTM_H];
#pragma unroll
    for (int j = 0; j < LSTM_H; ++j) {
      float a = fc1b[j];
#pragma unroll
      for (int k = 0; k < LSTM_H; ++k) a += fc1W[j * LSTM_H + k] * h1[k];
      r1[j] = fmaxf(a, 0.0f);
    }
    // --- layers 2, 3 ---
    lstm_cell(Wih2, Whh2, bih2, bhh2, r1, h2, c2);
    lstm_cell(Wih3, Whh3, bih3, bhh3, h2, h3, c3);
    // --- fc2 + sigmoid -> output ---
#pragma unroll
    for (int j = 0; j < LSTM_H; ++j) {
      float a = fc2b[j];
#pragma unroll
      for (int k = 0; k < LSTM_H; ++k) a += fc2W[j * LSTM_H + k] * h3[k];
      orow[j] = sigmoid_f(a);
    }
    xq += 5;
    orow += LSTM_H;
  }
}

// ---------------------------------------------------------------------------
extern "C" void kernel_launch(void* const* d_in, const int* in_sizes, int n_in,
                              void* d_out, int out_size, void* d_ws, size_t ws_size,
                              hipStream_t stream) {
  (void)n_in; (void)out_size; (void)ws_size;
  const float* x    = (const float*)d_in[0];
  const float* Wih0 = (const float*)d_in[1];
  const float* Whh0 = (const float*)d_in[2];
  const float* bih0 = (const float*)d_in[3];
  const float* bhh0 = (const float*)d_in[4];
  const float* Wih1 = (const float*)d_in[5];
  const float* Whh1 = (const float*)d_in[6];
  const float* bih1 = (const float*)d_in[7];
  const float* bhh1 = (const float*)d_in[8];
  const float* fc1W = (const float*)d_in[9];
  const float* fc1b = (const float*)d_in[10];
  const float* Wih2 = (const float*)d_in[11];
  const float* Whh2 = (const float*)d_in[12];
  const float* bih2 = (const float*)d_in[13];
  const float* bhh2 = (const float*)d_in[14];
  const float* Wih3 = (const float*)d_in[15];
  const float* Whh3 = (const float*)d_in[16];
  const float* bih3 = (const float*)d_in[17];
  const float* bhh3 = (const float*)d_in[18];
  const float* fc2W = (const float*)d_in[19];
  const float* fc2b = (const float*)d_in[20];

  const int M   = in_sizes[0] / DIN;   // B*T = 524288
  const int T   = 1024;
  const int Bsz = M / T;               // 512

  float* xg0 = (float*)d_ws;           // M * 20 floats = 40 MiB scratch

  // 16 rows per wave, 8 waves per block -> 128 rows/block; M % 128 == 0.
  xg_gemm_kernel<<<dim3(M / 128), dim3(256), 0, stream>>>(x, Wih0, bih0, bhh0, xg0);

  lstm_chain_kernel<<<dim3((Bsz + 255) / 256), dim3(256), 0, stream>>>(
      xg0, Whh0, Wih1, Whh1, bih1, bhh1, fc1W, fc1b,
      Wih2, Whh2, bih2, bhh2, Wih3, Whh3, bih3, bhh3, fc2W, fc2b,
      (float*)d_out, Bsz, T);
}